// GATEncoder_54571854463792
// MI455X (gfx1250) — compile-verified
//
#include <hip/hip_runtime.h>

#define N_NODES 100000
#define IN_DIM 128
#define HID 128
#define GAT_HEADS 8
#define NEG_SLOPE 0.2f

typedef float v2f __attribute__((ext_vector_type(2)));
typedef float v4f __attribute__((ext_vector_type(4)));
typedef float v8f __attribute__((ext_vector_type(8)));

// ---------------------------------------------------------------------------
// GEMM: H[nRows,128] = X[nRows,128] @ W[128,128]   (row-major, fp32 WMMA)
// One wave computes one 16x16 output tile; K-loop of 32 x V_WMMA_F32_16X16X4_F32.
// A frag (16x4 f32, 2 VGPR): lanes 0-15 hold K={0,1}, lanes 16-31 hold K={2,3}.
// B frag (4x16 f32, 2 VGPR): v0 = row K(+0) lanes0-15 / K(+2) lanes16-31, v1 = +1.
// C/D (16x16 f32, 8 VGPR): VGPR v -> row v (lanes 0-15) / row v+8 (lanes 16-31).
// ---------------------------------------------------------------------------
__global__ void __launch_bounds__(256) gemm128_wmma(const float* __restrict__ X,
                                                    const float* __restrict__ W,
                                                    float* __restrict__ H) {
  const int lane  = threadIdx.x & 31;
  const int wave  = blockIdx.x * (blockDim.x >> 5) + (threadIdx.x >> 5);
  const int tileM = wave >> 3;          // HID/16 == 8 N-tiles
  const int tileN = wave & 7;
  const int r     = lane & 15;
  const int khalf = lane >> 4;          // 0 or 1
  const int row   = tileM * 16 + r;
  const int col   = tileN * 16 + r;
  const float* __restrict__ xrow = X + (size_t)row * IN_DIM;

  v8f c = {0.f, 0.f, 0.f, 0.f, 0.f, 0.f, 0.f, 0.f};
#pragma unroll 4
  for (int k = 0; k < IN_DIM; k += 4) {
    const int kk = k + 2 * khalf;
    v2f a = *(const v2f*)(xrow + kk);
    v2f b;
    b.x = W[(kk    ) * HID + col];
    b.y = W[(kk + 1) * HID + col];
    c = __builtin_amdgcn_wmma_f32_16x16x4_f32(false, a, false, b, (short)0, c,
                                              false, false);
  }
  const int mbase = tileM * 16 + khalf * 8;
#pragma unroll
  for (int v = 0; v < 8; ++v)
    H[(size_t)(mbase + v) * HID + tileN * 16 + r] = c[v];
}

// ---------------------------------------------------------------------------
__global__ void fill_kernel(float* __restrict__ p, float v, int n) {
  int i = blockIdx.x * blockDim.x + threadIdx.x;
  if (i < n) p[i] = v;
}

__global__ void fill4_kernel(v4f* __restrict__ p, float v, int n4) {
  int i = blockIdx.x * blockDim.x + threadIdx.x;
  if (i < n4) { v4f t = {v, v, v, v}; p[i] = t; }
}

// per-(node,head) attention logits, layer 1 (8 heads x 16 ch)
__global__ void alpha1_kernel(const float* __restrict__ H,
                              const float* __restrict__ avS,
                              const float* __restrict__ avD,
                              float* __restrict__ outS,
                              float* __restrict__ outD) {
  int i = blockIdx.x * blockDim.x + threadIdx.x;   // node*8 + head
  if (i >= N_NODES * GAT_HEADS) return;
  int head = i & 7, node = i >> 3;
  const float* h  = H   + (size_t)node * HID + head * 16;
  const float* as = avS + head * 16;
  const float* ad = avD + head * 16;
  float s = 0.f, d = 0.f;
#pragma unroll
  for (int c = 0; c < 16; ++c) { float hv = h[c]; s += hv * as[c]; d += hv * ad[c]; }
  outS[i] = s; outD[i] = d;
}

// per-node attention logits, layer 2 (1 head x 128 ch), one wave per node
__global__ void alpha2_kernel(const float* __restrict__ H,
                              const float* __restrict__ avS,
                              const float* __restrict__ avD,
                              float* __restrict__ outS,
                              float* __restrict__ outD) {
  int wave = blockIdx.x * (blockDim.x >> 5) + (threadIdx.x >> 5);
  int lane = threadIdx.x & 31;
  if (wave >= N_NODES) return;
  float s = 0.f, d = 0.f;
#pragma unroll
  for (int c = lane; c < HID; c += 32) {
    float hv = H[(size_t)wave * HID + c];
    s += hv * avS[c]; d += hv * avD[c];
  }
#pragma unroll
  for (int off = 16; off > 0; off >>= 1) {
    s += __shfl_xor(s, off, 32);
    d += __shfl_xor(d, off, 32);
  }
  if (lane == 0) { outS[wave] = s; outD[wave] = d; }
}

// ---------------------------------------------------------------------------
__device__ __forceinline__ void edge_nodes(const int* __restrict__ ei, int e,
                                           int nEdges, int& s, int& d) {
  if (e < nEdges) { s = ei[e]; d = ei[nEdges + e]; }
  else            { s = d = e - nEdges; }          // appended self-loops
}

__device__ __forceinline__ float leaky(float x) {
  return x > 0.f ? x : NEG_SLOPE * x;
}

// monotone float atomic max via int-max / uint-min (init must be -inf)
__device__ __forceinline__ void atomicMaxFloat(float* addr, float val) {
  if (__float_as_int(val) >= 0)
    atomicMax((int*)addr, __float_as_int(val));
  else
    atomicMin((unsigned int*)addr, __float_as_uint(val));
}

template <int H>
__global__ void edge_max_kernel(const int* __restrict__ ei, int nEdges, int totE,
                                const float* __restrict__ aS,
                                const float* __restrict__ aD,
                                float* __restrict__ M) {
  int i = blockIdx.x * blockDim.x + threadIdx.x;
  if (i >= totE * H) return;
  int h = i % H, e = i / H, s, d;
  edge_nodes(ei, e, nEdges, s, d);
  atomicMaxFloat(&M[d * H + h], leaky(aS[s * H + h] + aD[d * H + h]));
}

template <int H>
__global__ void edge_sum_kernel(const int* __restrict__ ei, int nEdges, int totE,
                                const float* __restrict__ aS,
                                const float* __restrict__ aD,
                                const float* __restrict__ M,
                                float* __restrict__ S) {
  int i = blockIdx.x * blockDim.x + threadIdx.x;
  if (i >= totE * H) return;
  int h = i % H, e = i / H, s, d;
  edge_nodes(ei, e, nEdges, s, d);
  float x = leaky(aS[s * H + h] + aD[d * H + h]);
  atomicAdd(&S[d * H + h], expf(x - M[d * H + h]));
}

// one thread per (edge, 4-channel group): b128 gather + 4 f32 atomic adds.
// A float4 group never straddles a 16-channel head boundary.
template <int H>
__global__ void edge_aggr_kernel(const int* __restrict__ ei, int nEdges, int totE,
                                 const float* __restrict__ aS,
                                 const float* __restrict__ aD,
                                 const float* __restrict__ M,
                                 const float* __restrict__ S,
                                 const float* __restrict__ Hf,
                                 float* __restrict__ acc) {
  int i = blockIdx.x * blockDim.x + threadIdx.x;   // edge*32 + group
  if (i >= totE * (HID / 4)) return;
  int g = i & 31, e = i >> 5, s, d;
  const int c = g * 4;
  edge_nodes(ei, e, nEdges, s, d);
  const int h = c / (HID / H);
  float x = leaky(aS[s * H + h] + aD[d * H + h]);
  float w = expf(x - M[d * H + h]) / S[d * H + h];
  v4f hv = *(const v4f*)(Hf + (size_t)s * HID + c);
  float* a = acc + (size_t)d * HID + c;
  atomicAdd(a + 0, hv.x * w);
  atomicAdd(a + 1, hv.y * w);
  atomicAdd(a + 2, hv.z * w);
  atomicAdd(a + 3, hv.w * w);
}

// out = elu(acc + bias), vectorized x4
__global__ void finalize_elu_kernel(const v4f* __restrict__ acc,
                                    const float* __restrict__ bias,
                                    v4f* __restrict__ out) {
  int i = blockIdx.x * blockDim.x + threadIdx.x;   // over N*32 float4 groups
  if (i >= N_NODES * (HID / 4)) return;
  const int c = (i & 31) * 4;
  v4f v = acc[i];
  v.x += bias[c + 0]; v.y += bias[c + 1]; v.z += bias[c + 2]; v.w += bias[c + 3];
  v.x = v.x > 0.f ? v.x : (expf(v.x) - 1.f);
  v.y = v.y > 0.f ? v.y : (expf(v.y) - 1.f);
  v.z = v.z > 0.f ? v.z : (expf(v.z) - 1.f);
  v.w = v.w > 0.f ? v.w : (expf(v.w) - 1.f);
  out[i] = v;
}

__global__ void bias_kernel(v4f* __restrict__ out, const float* __restrict__ bias) {
  int i = blockIdx.x * blockDim.x + threadIdx.x;
  if (i >= N_NODES * (HID / 4)) return;
  const int c = (i & 31) * 4;
  v4f v = out[i];
  v.x += bias[c + 0]; v.y += bias[c + 1]; v.z += bias[c + 2]; v.w += bias[c + 3];
  out[i] = v;
}

// ---------------------------------------------------------------------------
extern "C" void kernel_launch(void* const* d_in, const int* in_sizes, int n_in,
                              void* d_out, int out_size, void* d_ws, size_t ws_size,
                              hipStream_t stream) {
  const float* x   = (const float*)d_in[0];
  const int*   ei  = (const int*)d_in[1];
  const float* W1  = (const float*)d_in[2];
  const float* as1 = (const float*)d_in[3];
  const float* ad1 = (const float*)d_in[4];
  const float* b1  = (const float*)d_in[5];
  const float* W2  = (const float*)d_in[6];
  const float* as2 = (const float*)d_in[7];
  const float* ad2 = (const float*)d_in[8];
  const float* b2  = (const float*)d_in[9];

  const int nEdges = in_sizes[1] / 2;
  const int totE   = nEdges + N_NODES;

  float* A   = (float*)d_ws;                        // [N,128] features
  float* B   = A  + (size_t)N_NODES * HID;          // [N,128] accumulator / h2
  float* aS  = B  + (size_t)N_NODES * HID;          // [N,8]
  float* aD  = aS + (size_t)N_NODES * GAT_HEADS;
  float* Mx  = aD + (size_t)N_NODES * GAT_HEADS;
  float* Sx  = Mx + (size_t)N_NODES * GAT_HEADS;
  float* out = (float*)d_out;

  const int TB = 256;
  auto cdiv = [](long long a, long long b) { return (unsigned)((a + b - 1) / b); };
  const unsigned gNode4 = cdiv((long long)N_NODES * (HID / 4), TB); // N*32 groups
  const unsigned gGemm  = (N_NODES / 16) * (HID / 16) / (TB / 32);
  const unsigned gAggr  = cdiv((long long)totE * (HID / 4), TB);
  const float NEG_INF = -__builtin_huge_valf();

  // ================= layer 1 (8 heads) =================
  gemm128_wmma<<<gGemm, TB, 0, stream>>>(x, W1, A);
  alpha1_kernel<<<cdiv(N_NODES * GAT_HEADS, TB), TB, 0, stream>>>(A, as1, ad1, aS, aD);
  fill_kernel<<<cdiv(N_NODES * GAT_HEADS, TB), TB, 0, stream>>>(Mx, NEG_INF, N_NODES * GAT_HEADS);
  fill_kernel<<<cdiv(N_NODES * GAT_HEADS, TB), TB, 0, stream>>>(Sx, 0.f, N_NODES * GAT_HEADS);
  fill4_kernel<<<gNode4, TB, 0, stream>>>((v4f*)B, 0.f, N_NODES * (HID / 4));
  edge_max_kernel<GAT_HEADS><<<cdiv((long long)totE * GAT_HEADS, TB), TB, 0, stream>>>(
      ei, nEdges, totE, aS, aD, Mx);
  edge_sum_kernel<GAT_HEADS><<<cdiv((long long)totE * GAT_HEADS, TB), TB, 0, stream>>>(
      ei, nEdges, totE, aS, aD, Mx, Sx);
  edge_aggr_kernel<GAT_HEADS><<<gAggr, TB, 0, stream>>>(
      ei, nEdges, totE, aS, aD, Mx, Sx, A, B);
  finalize_elu_kernel<<<gNode4, TB, 0, stream>>>((const v4f*)B, b1, (v4f*)A);

  // ================= layer 2 (1 head) =================
  gemm128_wmma<<<gGemm, TB, 0, stream>>>(A, W2, B);          // B = h2
  alpha2_kernel<<<cdiv((long long)N_NODES * 32, TB), TB, 0, stream>>>(B, as2, ad2, aS, aD);
  fill_kernel<<<cdiv(N_NODES, TB), TB, 0, stream>>>(Mx, NEG_INF, N_NODES);
  fill_kernel<<<cdiv(N_NODES, TB), TB, 0, stream>>>(Sx, 0.f, N_NODES);
  fill4_kernel<<<gNode4, TB, 0, stream>>>((v4f*)out, 0.f, N_NODES * (HID / 4));
  edge_max_kernel<1><<<cdiv(totE, TB), TB, 0, stream>>>(ei, nEdges, totE, aS, aD, Mx);
  edge_sum_kernel<1><<<cdiv(totE, TB), TB, 0, stream>>>(ei, nEdges, totE, aS, aD, Mx, Sx);
  edge_aggr_kernel<1><<<gAggr, TB, 0, stream>>>(
      ei, nEdges, totE, aS, aD, Mx, Sx, B, out);
  bias_kernel<<<gNode4, TB, 0, stream>>>((v4f*)out, b2);
}